// GritMessagePassingLayer_42863773614510
// MI455X (gfx1250) — compile-verified
//
#include <hip/hip_runtime.h>
#include <hip/hip_bf16.h>

// Problem constants (match reference)
#define NN 50000
#define EE 600000
#define DD 128
#define HH 8
#define HDD 16
#define EPS_BN 1e-5f
#define CLAMPV 5.0f
// monotonic encoding of -CLAMPV (-5.0f = 0xC0A00000, sign set -> ~u)
#define SMAX_INIT_ENC 0x3F5FFFFFu

typedef __attribute__((ext_vector_type(2))) float v2f;
typedef __attribute__((ext_vector_type(8))) float v8f;

// ---- monotonic float<->uint encoding for atomicMax on floats ----
__device__ __forceinline__ unsigned fenc(float f) {
  unsigned u = __float_as_uint(f);
  return (u & 0x80000000u) ? ~u : (u | 0x80000000u);
}
__device__ __forceinline__ float fdec(unsigned e) {
  return (e & 0x80000000u) ? __uint_as_float(e & 0x7fffffffu)
                           : __uint_as_float(~e);
}

// =====================================================================
// Generic fp32 WMMA GEMM:  C[M,Ncols] = A[M,K] @ W[Ncols,K]^T (+bias)(+R)(relu?)
// block = 256 threads (8 waves); wave w -> 16x16 tile at cols blockIdx.y*128+w*16
// A tile (16 x K) staged in LDS with +4 padding (conflict-free v2f reads).
// =====================================================================
template <int K, bool RELU>
__global__ __launch_bounds__(256) void gemm_wmma(
    const float* __restrict__ A, const float* __restrict__ W,
    const float* __restrict__ bias, const float* __restrict__ R,
    float* __restrict__ C, int M, int Ncols) {
  __shared__ float As[16 * (K + 4)];
  const int tid  = threadIdx.x;
  const int wave = tid >> 5;
  const int lane = tid & 31;
  const int m    = lane & 15;   // M (for A frag) and N (for B frag)
  const int g    = lane >> 4;   // K-half selector
  const long tileM   = (long)blockIdx.x * 16;
  const int  colBase = blockIdx.y * 128;

  // cooperative A-tile load into LDS
  for (int idx = tid; idx < 16 * K; idx += 256) {
    int r = idx / K, k = idx % K;
    As[r * (K + 4) + k] = A[(tileM + r) * (long)K + k];
  }
  __syncthreads();

  const float* Wp = W + (long)(colBase + wave * 16 + m) * K;
  v8f acc = {};
#pragma unroll
  for (int k0 = 0; k0 < K; k0 += 4) {
    v2f a = *(const v2f*)&As[m * (K + 4) + k0 + 2 * g];
    v2f b = *(const v2f*)&Wp[k0 + 2 * g];
    acc = __builtin_amdgcn_wmma_f32_16x16x4_f32(false, a, false, b, (short)0,
                                                acc, false, false);
  }

  const int col = colBase + wave * 16 + m;
  const float bv = bias ? bias[col] : 0.0f;
#pragma unroll
  for (int r = 0; r < 8; ++r) {
    long row = tileM + r + 8 * g;
    float v = acc[r] + bv;
    if (R) v += R[row * (long)Ncols + col];
    if (RELU) v = fmaxf(v, 0.0f);
    C[row * (long)Ncols + col] = v;
  }
}

// =====================================================================
// fill kernel (for smax init with encoded -CLAMP)
// =====================================================================
__global__ void fill_u32(unsigned* p, long n, unsigned v) {
  long i = (long)blockIdx.x * blockDim.x + threadIdx.x;
  long s = (long)gridDim.x * blockDim.x;
  for (; i < n; i += s) p[i] = v;
}

// =====================================================================
// Edge nonlinearity: conn = relu(signed_sqrt((Q[dst]+K[src])*Ew)+Eb)
// + per-head attention score (clamped) + segment-max + degree count.
// one wave per edge, 4 channels per lane.
// =====================================================================
__global__ __launch_bounds__(256) void edge_conn(
    const float* __restrict__ Qh, const float* __restrict__ Kh,
    const float* EwB, const float* __restrict__ EbB,
    const float* __restrict__ Aw, const int* __restrict__ eidx, float* conn,
    float* __restrict__ score, unsigned* __restrict__ smax,
    float* __restrict__ deg) {
  long gid = (long)blockIdx.x * blockDim.x + threadIdx.x;
  long e = gid >> 5;
  int lane = (int)(gid & 31);
  if (e >= EE) return;
  int dst = eidx[e];
  int src = eidx[EE + e];
  int d0 = lane * 4;

  float4 qv = *(const float4*)(Qh + (long)dst * DD + d0);
  float4 kv = *(const float4*)(Kh + (long)src * DD + d0);
  float4 ew = *(const float4*)(EwB + e * DD + d0);
  float4 eb = *(const float4*)(EbB + e * DD + d0);
  float qk[4] = {qv.x + kv.x, qv.y + kv.y, qv.z + kv.z, qv.w + kv.w};
  float ewv[4] = {ew.x, ew.y, ew.z, ew.w};
  float ebv[4] = {eb.x, eb.y, eb.z, eb.w};
  float c[4];
#pragma unroll
  for (int i = 0; i < 4; ++i) {
    float t = qk[i] * ewv[i];
    float s = sqrtf(fabsf(t));
    s = (t > 0.f) ? s : ((t < 0.f) ? -s : 0.f);
    c[i] = fmaxf(s + ebv[i], 0.f);
  }
  *(float4*)(conn + e * DD + d0) = make_float4(c[0], c[1], c[2], c[3]);

  // score[e,h] = sum_d conn[e,h,d] * Aw[d,h]
  int h = lane >> 2;
  int dl = (lane & 3) * 4;
  float part = c[0] * Aw[(dl + 0) * HH + h] + c[1] * Aw[(dl + 1) * HH + h] +
               c[2] * Aw[(dl + 2) * HH + h] + c[3] * Aw[(dl + 3) * HH + h];
  part += __shfl_xor(part, 1, 32);
  part += __shfl_xor(part, 2, 32);
  if ((lane & 3) == 0) {
    float sc = fminf(fmaxf(part, -CLAMPV), CLAMPV);
    score[e * HH + h] = sc;
    atomicMax(smax + (long)dst * HH + h, fenc(sc));
  }
  if (lane == 0) atomicAdd(deg + src, 1.0f);
}

// ex = exp(score - smax[dst]); den[dst] += ex
__global__ void edge_exp(const int* __restrict__ eidx, float* __restrict__ score,
                         const unsigned* __restrict__ smax,
                         float* __restrict__ den) {
  long idx = (long)blockIdx.x * blockDim.x + threadIdx.x;
  if (idx >= (long)EE * HH) return;
  long e = idx >> 3;
  int h = (int)(idx & 7);
  int dst = eidx[e];
  float m = fdec(smax[(long)dst * HH + h]);
  float ex = expf(score[idx] - m);
  score[idx] = ex;
  atomicAdd(den + (long)dst * HH + h, ex);
}

// normalized-score weighted scatter of V[src] and Eo[e] into dst rows
__global__ __launch_bounds__(256) void edge_agg(
    const int* __restrict__ eidx, const float* __restrict__ score,
    const float* __restrict__ den, const float* __restrict__ Vh,
    const float* __restrict__ Eo, float* __restrict__ agg,
    float* __restrict__ rowAcc) {
  long gid = (long)blockIdx.x * blockDim.x + threadIdx.x;
  long e = gid >> 5;
  int lane = (int)(gid & 31);
  if (e >= EE) return;
  int dst = eidx[e];
  int src = eidx[EE + e];
  int h = lane >> 2;
  int d0 = lane * 4;
  float s = score[e * HH + h] / (den[(long)dst * HH + h] + 1e-16f);
  float4 v = *(const float4*)(Vh + (long)src * DD + d0);
  float4 eo = *(const float4*)(Eo + e * DD + d0);
  float* ap = agg + (long)dst * DD + d0;
  float* rp = rowAcc + (long)dst * DD + d0;
  atomicAdd(ap + 0, v.x * s);
  atomicAdd(ap + 1, v.y * s);
  atomicAdd(ap + 2, v.z * s);
  atomicAdd(ap + 3, v.w * s);
  atomicAdd(rp + 0, eo.x * s);
  atomicAdd(rp + 1, eo.y * s);
  atomicAdd(rp + 2, eo.z * s);
  atomicAdd(rp + 3, eo.w * s);
}

// rowV = einsum(rowAcc, BW); h = (agg+rowV)*deg_coef0 + (agg+rowV)*log1p(deg)*deg_coef1
__global__ void node_combine(const float* __restrict__ agg,
                             const float* __restrict__ rowAcc,
                             const float* __restrict__ BW,
                             const float* __restrict__ deg,
                             const float* __restrict__ deg_coef,
                             float* __restrict__ hmid) {
  long idx = (long)blockIdx.x * blockDim.x + threadIdx.x;
  if (idx >= (long)NN * DD) return;
  long n = idx >> 7;
  int ch = (int)(idx & 127);
  int h = ch >> 4;
  int c = ch & 15;
  const float* ra = rowAcc + n * DD + h * HDD;
  float rv = 0.f;
#pragma unroll
  for (int d = 0; d < HDD; ++d) rv += ra[d] * BW[d * DD + h * HDD + c];
  float val = agg[idx] + rv;
  float ld = log1pf(deg[n]);
  float a = deg_coef[ch * 2 + 0], b = deg_coef[ch * 2 + 1];
  hmid[idx] = val * a + val * ld * b;
}

// ---- BatchNorm (training mode, biased var): two-pass column stats ----
__global__ void bn_stats(const float* __restrict__ X, long M,
                         float* __restrict__ sum, float* __restrict__ sumsq) {
  long tid = (long)blockIdx.x * blockDim.x + threadIdx.x;
  int col = (int)(tid & 127);
  long row = tid >> 7;
  long rstride = ((long)gridDim.x * blockDim.x) >> 7;
  float s = 0.f, ss = 0.f;
  for (long r = row; r < M; r += rstride) {
    float v = X[r * DD + col];
    s += v;
    ss += v * v;
  }
  atomicAdd(sum + col, s);
  atomicAdd(sumsq + col, ss);
}

__global__ void bn_norm(const float* __restrict__ X, float* __restrict__ Y,
                        const float* __restrict__ sum,
                        const float* __restrict__ sumsq,
                        const float* __restrict__ gamma,
                        const float* __restrict__ beta, long M) {
  long idx = (long)blockIdx.x * blockDim.x + threadIdx.x;
  if (idx >= M * DD) return;
  int col = (int)(idx & 127);
  float mu = sum[col] / (float)M;
  float var = sumsq[col] / (float)M - mu * mu;
  float inv = rsqrtf(var + EPS_BN);
  Y[idx] = (X[idx] - mu) * inv * gamma[col] + beta[col];
}

// =====================================================================
// host side
// =====================================================================
extern "C" void kernel_launch(void* const* d_in, const int* in_sizes, int n_in,
                              void* d_out, int out_size, void* d_ws,
                              size_t ws_size, hipStream_t stream) {
  (void)in_sizes; (void)n_in; (void)out_size; (void)ws_size;
  // inputs in setup_inputs() order (params dict flattened in insertion order)
  const float* x         = (const float*)d_in[0];
  const float* edge_attr = (const float*)d_in[1];
  const int*   eidx      = (const int*)d_in[2];
  const float* WQ  = (const float*)d_in[3];
  const float* WK  = (const float*)d_in[4];
  const float* WV  = (const float*)d_in[5];
  const float* WEw = (const float*)d_in[6];
  const float* WEb = (const float*)d_in[7];
  const float* bEb = (const float*)d_in[8];
  const float* WEo = (const float*)d_in[9];
  const float* bEo = (const float*)d_in[10];
  const float* Aw  = (const float*)d_in[11];
  const float* BW  = (const float*)d_in[12];
  const float* deg_coef = (const float*)d_in[13];
  const float* WOh = (const float*)d_in[14];
  const float* bOh = (const float*)d_in[15];
  const float* WOe = (const float*)d_in[16];
  const float* bOe = (const float*)d_in[17];
  const float* g1h = (const float*)d_in[18];
  const float* b1h = (const float*)d_in[19];
  const float* g1e = (const float*)d_in[20];
  const float* b1e = (const float*)d_in[21];
  const float* W1  = (const float*)d_in[22];
  const float* b1  = (const float*)d_in[23];
  const float* W2  = (const float*)d_in[24];
  const float* b2  = (const float*)d_in[25];
  const float* g2h = (const float*)d_in[26];
  const float* b2h = (const float*)d_in[27];

  float* outN = (float*)d_out;                       // [N,128]
  float* outE = (float*)d_out + (size_t)NN * DD;     // [E,128]

  // workspace carve-up
  float* ws = (float*)d_ws;
  size_t off = 0;
  auto alloc = [&](size_t n) { float* p = ws + off; off += n; return p; };
  float* Qh    = alloc((size_t)NN * DD);
  float* Kh    = alloc((size_t)NN * DD);
  float* Vh    = alloc((size_t)NN * DD);
  float* hmid  = alloc((size_t)NN * DD);
  float* h_res = alloc((size_t)NN * DD);
  float* h1    = alloc((size_t)NN * DD);
  float* hsum  = alloc((size_t)NN * DD);
  float* ffn   = alloc((size_t)NN * 2 * DD);
  // zero-initialized block (single memset)
  float* agg    = alloc((size_t)NN * DD);
  float* rowAcc = alloc((size_t)NN * DD);
  float* deg    = alloc((size_t)NN);
  float* den    = alloc((size_t)NN * HH);
  float* stats  = alloc(6 * DD);  // 3 BNs x (sum, sumsq)
  size_t zeroFloats = (size_t)NN * DD * 2 + NN + (size_t)NN * HH + 6 * DD;
  unsigned* smax = (unsigned*)alloc((size_t)NN * HH);
  float* score = alloc((size_t)EE * HH);
  float* EwB = alloc((size_t)EE * DD);  // becomes conn (in-place)
  float* EbB = alloc((size_t)EE * DD);
  float* EoB = alloc((size_t)EE * DD);

  float* sum1h = stats + 0 * DD; float* sq1h = stats + 1 * DD;
  float* sum2h = stats + 2 * DD; float* sq2h = stats + 3 * DD;
  float* sum1e = stats + 4 * DD; float* sq1e = stats + 5 * DD;

  (void)hipMemsetAsync(agg, 0, zeroFloats * sizeof(float), stream);
  fill_u32<<<1563, 256, 0, stream>>>(smax, (long)NN * HH, SMAX_INIT_ENC);

  dim3 blk(256);
  // node projections Q,K,V
  gemm_wmma<128, false><<<dim3(NN / 16, 1), blk, 0, stream>>>(x, WQ, nullptr, nullptr, Qh, NN, DD);
  gemm_wmma<128, false><<<dim3(NN / 16, 1), blk, 0, stream>>>(x, WK, nullptr, nullptr, Kh, NN, DD);
  gemm_wmma<128, false><<<dim3(NN / 16, 1), blk, 0, stream>>>(x, WV, nullptr, nullptr, Vh, NN, DD);
  // edge projections Ew, Eb(+bias)
  gemm_wmma<128, false><<<dim3(EE / 16, 1), blk, 0, stream>>>(edge_attr, WEw, nullptr, nullptr, EwB, EE, DD);
  gemm_wmma<128, false><<<dim3(EE / 16, 1), blk, 0, stream>>>(edge_attr, WEb, bEb, nullptr, EbB, EE, DD);
  // conn + score + segment-max + degree
  edge_conn<<<(EE * 32) / 256, blk, 0, stream>>>(Qh, Kh, EwB, EbB, Aw, eidx,
                                                 EwB, score, smax, deg);
  // Eo = conn @ WEo^T + bEo
  gemm_wmma<128, false><<<dim3(EE / 16, 1), blk, 0, stream>>>(EwB, WEo, bEo, nullptr, EoB, EE, DD);
  // softmax pieces
  edge_exp<<<(EE * HH) / 256, blk, 0, stream>>>(eidx, score, smax, den);
  edge_agg<<<(EE * 32) / 256, blk, 0, stream>>>(eidx, score, den, Vh, EoB, agg, rowAcc);
  // per-node combine (BW einsum + degree scaler)
  node_combine<<<(NN * DD) / 256, blk, 0, stream>>>(agg, rowAcc, BW, deg, deg_coef, hmid);
  // h_res = x + hmid @ WOh^T + bOh ; BN1h -> h1
  gemm_wmma<128, false><<<dim3(NN / 16, 1), blk, 0, stream>>>(hmid, WOh, bOh, x, h_res, NN, DD);
  bn_stats<<<512, blk, 0, stream>>>(h_res, NN, sum1h, sq1h);
  bn_norm<<<(NN * DD) / 256, blk, 0, stream>>>(h_res, h1, sum1h, sq1h, g1h, b1h, NN);
  // FFN: relu(h1 @ W1^T + b1) @ W2^T + b2 + h1 ; BN2h -> outN
  gemm_wmma<128, true><<<dim3(NN / 16, 2), blk, 0, stream>>>(h1, W1, b1, nullptr, ffn, NN, 2 * DD);
  gemm_wmma<256, false><<<dim3(NN / 16, 1), blk, 0, stream>>>(ffn, W2, b2, h1, hsum, NN, DD);
  bn_stats<<<512, blk, 0, stream>>>(hsum, NN, sum2h, sq2h);
  bn_norm<<<(NN * DD) / 256, blk, 0, stream>>>(hsum, outN, sum2h, sq2h, g2h, b2h, NN);
  // e_out = edge_attr + Eo @ WOe^T + bOe ; BN1e (in-place into outE)
  gemm_wmma<128, false><<<dim3(EE / 16, 1), blk, 0, stream>>>(EoB, WOe, bOe, edge_attr, outE, EE, DD);
  bn_stats<<<512, blk, 0, stream>>>(outE, EE, sum1e, sq1e);
  bn_norm<<<(EE * DD) / 256, blk, 0, stream>>>(outE, outE, sum1e, sq1e, g1e, b1e, EE);
}